// EncoderMinLSTMGNN_24962349924441
// MI455X (gfx1250) — compile-verified
//
#include <hip/hip_runtime.h>
#include <math.h>

#define N_NODES 512
#define SEQ 64
#define DIN 128
#define HEADS 4
#define DOUT 128
#define NEDGE 4096
#define NEG_SLOPE 0.2f
#define LN_EPS 1e-5f

typedef __attribute__((ext_vector_type(2))) float v2f;
typedef __attribute__((ext_vector_type(8))) float v8f;

__device__ __forceinline__ float leaky(float v) { return v > 0.f ? v : NEG_SLOPE * v; }

// ---------------- preprocess: keep ch0, log10(rest+1), NaN->0 ----------------
__global__ void k_preprocess(const float* __restrict__ x, float* __restrict__ xp, int total) {
    int i = blockIdx.x * blockDim.x + threadIdx.x;
    if (i >= total) return;
    int d = i & (DIN - 1);
    float v = x[i];
    if (d != 0) v = log10f(v + 1.0f);
    if (v != v) v = 0.0f;
    xp[i] = v;
}

// ---------------- deterministic CSR build (one block, 512 threads) ----------------
__global__ void k_build_csr(const int* __restrict__ dst, int* __restrict__ rowptr,
                            int* __restrict__ eidx) {
    __shared__ int deg[N_NODES];
    __shared__ int rp[N_NODES + 1];
    int v = threadIdx.x;
    int cnt = 0;
    for (int e = 0; e < NEDGE; ++e) cnt += (dst[e] == v);
    deg[v] = cnt;
    __syncthreads();
    if (v == 0) {
        int acc = 0;
        for (int i = 0; i < N_NODES; ++i) { rp[i] = acc; acc += deg[i]; }
        rp[N_NODES] = acc;
    }
    __syncthreads();
    int pos = rp[v];
    for (int e = 0; e < NEDGE; ++e)
        if (dst[e] == v) eidx[pos++] = e;   // stable (ascending e) -> deterministic
    rowptr[v] = rp[v];
    if (v == 0) rowptr[N_NODES] = rp[N_NODES];
}

// ---------------- fp32 WMMA GEMM: C[M,Nc] = A[M,K] @ W[K,Nc] + bias ----------------
// grid = (M/(16*MT), Nc/128), block = 256 (8 wave32).
// Each wave owns a 16-col slice and MT stacked 16x16 M-tiles sharing each B frag.
// A tile [16*MT x K] staged in LDS via CDNA5 async global->LDS copies (ASYNCcnt).
template <int MT>
__global__ __launch_bounds__(256) void k_gemm_wmma(
    const float* __restrict__ A, const float* __restrict__ W,
    const float* __restrict__ bias, float* __restrict__ C,
    int M, int K, int Nc) {
    extern __shared__ float As[];           // [16*MT rows x K], dynamic LDS only
    const int tid = threadIdx.x;
    const int wave = tid >> 5;
    const int lane = tid & 31;
    const int rows = 16 * MT;
    const int mbase = blockIdx.x * rows;
    const int colbase = blockIdx.y * 128 + wave * 16;

    // ---- stage A tile into LDS with async b128 copies (per-lane addresses) ----
    const int kv = K >> 2;                  // float4 chunks per row
    for (int idx = tid; idx < rows * kv; idx += 256) {
        int r = idx / kv;
        int c4 = (idx - r * kv) << 2;
        unsigned loff = (unsigned)((r * K + c4) * 4);            // LDS byte offset
        const float* gp = &A[(size_t)(mbase + r) * K + c4];
        asm volatile("global_load_async_to_lds_b128 %0, %1, off"
                     :: "v"(loff), "v"(gp) : "memory");
    }
    asm volatile("s_wait_asynccnt 0x0" ::: "memory");
    __syncthreads();

    const int halflane = lane & 15;
    const int hi = lane >> 4;               // 0 or 1
    const int koff = hi * 2;                // A/B: lanes 16-31 hold K+2,K+3

    float bv = bias[colbase + halflane];
    v8f acc[MT];
#pragma unroll
    for (int t2 = 0; t2 < MT; ++t2) acc[t2] = (v8f){bv, bv, bv, bv, bv, bv, bv, bv};

    const float* Wp = &W[colbase + halflane];
    for (int k0 = 0; k0 < K; k0 += 4) {
        v2f b;
        b.x = Wp[(size_t)(k0 + koff) * Nc];
        b.y = Wp[(size_t)(k0 + koff + 1) * Nc];
#pragma unroll
        for (int t2 = 0; t2 < MT; ++t2) {
            v2f a = *(const v2f*)&As[(halflane + 16 * t2) * K + k0 + koff];
            acc[t2] = __builtin_amdgcn_wmma_f32_16x16x4_f32(false, a, false, b,
                                                            (short)0, acc[t2],
                                                            false, false);
        }
    }
#pragma unroll
    for (int t2 = 0; t2 < MT; ++t2) {
#pragma unroll
        for (int i = 0; i < 8; ++i) {       // C: VGPR i -> M=i (lanes 0-15) / M=8+i
            int m = 16 * t2 + i + hi * 8;
            C[(size_t)(mbase + m) * Nc + colbase + halflane] = acc[t2][i];
        }
    }
}

// ---------------- layer-1 edge logits: ex1[e,s,h] = exp(attn1[h]·leakyrelu(fs[u]+fd[v])) ----------------
__global__ __launch_bounds__(256) void k_logits1(
    const float* __restrict__ fs, const float* __restrict__ fd,
    const int* __restrict__ src, const int* __restrict__ dst,
    const float* __restrict__ attn, float* __restrict__ ex) {
    int gid = blockIdx.x * 8 + (threadIdx.x >> 5);   // wave per (e,s)
    int lane = threadIdx.x & 31;
    int e = gid / SEQ;
    int s = gid - e * SEQ;
    int u = src[e], v = dst[e];
    const float4* pfs = (const float4*)&fs[((size_t)u * SEQ + s) * (HEADS * DIN)];
    const float4* pfd = (const float4*)&fd[((size_t)v * SEQ + s) * (HEADS * DIN)];
    const float4* pat = (const float4*)attn;
#pragma unroll
    for (int h = 0; h < HEADS; ++h) {
        int q = h * (DIN / 4) + lane;                // one float4 per lane
        float4 a = pfs[q], b = pfd[q], w = pat[q];
        float sum = leaky(a.x + b.x) * w.x + leaky(a.y + b.y) * w.y +
                    leaky(a.z + b.z) * w.z + leaky(a.w + b.w) * w.w;
        for (int off = 16; off; off >>= 1) sum += __shfl_xor(sum, off, 32);
        if (lane == 0) ex[((size_t)e * SEQ + s) * HEADS + h] = expf(sum);
    }
}

__global__ void k_den1(const float* __restrict__ ex, const int* __restrict__ rowptr,
                       const int* __restrict__ eidx, float* __restrict__ den) {
    int i = blockIdx.x * blockDim.x + threadIdx.x;   // over N*SEQ*HEADS
    if (i >= N_NODES * SEQ * HEADS) return;
    int h = i & (HEADS - 1);
    int t = i >> 2;
    int s = t & (SEQ - 1);
    int v = t >> 6;
    float acc = 0.f;
    int pe = rowptr[v + 1];
    for (int p = rowptr[v]; p < pe; ++p)
        acc += ex[((size_t)eidx[p] * SEQ + s) * HEADS + h];
    den[i] = acc;
}

// ---------------- layer-1 aggregation: g1[v,s,:] = sum alpha * fs1[u,s,:] ----------------
__global__ __launch_bounds__(256) void k_agg1(
    const float* __restrict__ fs, const float* __restrict__ ex,
    const float* __restrict__ den, const int* __restrict__ src,
    const int* __restrict__ rowptr, const int* __restrict__ eidx,
    float* __restrict__ g) {
    int vs = blockIdx.x;                   // v*SEQ + s
    int s = vs & (SEQ - 1);
    int v = vs >> 6;
    int t = threadIdx.x;
    int d0 = t, d1 = t + 256;
    int h0 = d0 >> 7, h1 = d1 >> 7;
    float rdn0 = 1.f / den[(size_t)vs * HEADS + h0];
    float rdn1 = 1.f / den[(size_t)vs * HEADS + h1];
    float acc0 = 0.f, acc1 = 0.f;
    int pe = rowptr[v + 1];
    for (int p = rowptr[v]; p < pe; ++p) {
        int e = eidx[p];
        int u = src[e];
        const float* frow = &fs[((size_t)u * SEQ + s) * (HEADS * DIN)];
        const float* exr = &ex[((size_t)e * SEQ + s) * HEADS];
        acc0 += (exr[h0] * rdn0) * frow[d0];
        acc1 += (exr[h1] * rdn1) * frow[d1];
    }
    float* grow = &g[(size_t)vs * (HEADS * DIN)];
    grow[d0] = acc0;
    grow[d1] = acc1;
}

// ---------------- layer-2 edge logits (1 head) ----------------
__global__ __launch_bounds__(256) void k_logits2(
    const float* __restrict__ fs, const float* __restrict__ fd,
    const int* __restrict__ src, const int* __restrict__ dst,
    const float* __restrict__ attn, float* __restrict__ ex) {
    int gid = blockIdx.x * 8 + (threadIdx.x >> 5);
    int lane = threadIdx.x & 31;
    int e = gid / SEQ;
    int s = gid - e * SEQ;
    int u = src[e], v = dst[e];
    const float4* pfs = (const float4*)&fs[((size_t)u * SEQ + s) * DIN];
    const float4* pfd = (const float4*)&fd[((size_t)v * SEQ + s) * DIN];
    const float4* pat = (const float4*)attn;
    float4 a = pfs[lane], b = pfd[lane], w = pat[lane];
    float sum = leaky(a.x + b.x) * w.x + leaky(a.y + b.y) * w.y +
                leaky(a.z + b.z) * w.z + leaky(a.w + b.w) * w.w;
    for (int off = 16; off; off >>= 1) sum += __shfl_xor(sum, off, 32);
    if (lane == 0) ex[(size_t)e * SEQ + s] = expf(sum);
}

__global__ void k_den2(const float* __restrict__ ex, const int* __restrict__ rowptr,
                       const int* __restrict__ eidx, float* __restrict__ den) {
    int i = blockIdx.x * blockDim.x + threadIdx.x;   // over N*SEQ
    if (i >= N_NODES * SEQ) return;
    int s = i & (SEQ - 1);
    int v = i >> 6;
    float acc = 0.f;
    int pe = rowptr[v + 1];
    for (int p = rowptr[v]; p < pe; ++p) acc += ex[(size_t)eidx[p] * SEQ + s];
    den[i] = acc;
}

__global__ __launch_bounds__(128) void k_agg2(
    const float* __restrict__ fs, const float* __restrict__ ex,
    const float* __restrict__ den, const int* __restrict__ src,
    const int* __restrict__ rowptr, const int* __restrict__ eidx,
    float* __restrict__ g) {
    int vs = blockIdx.x;
    int s = vs & (SEQ - 1);
    int v = vs >> 6;
    int d = threadIdx.x;
    float rdn = 1.f / den[vs];
    float acc = 0.f;
    int pe = rowptr[v + 1];
    for (int p = rowptr[v]; p < pe; ++p) {
        int e = eidx[p];
        int u = src[e];
        acc += (ex[(size_t)e * SEQ + s] * rdn) * fs[((size_t)u * SEQ + s) * DIN + d];
    }
    g[(size_t)vs * DIN + d] = acc;
}

// ---------------- residual + LayerNorm ----------------
__global__ __launch_bounds__(128) void k_res_ln(
    const float* __restrict__ xp, const float* __restrict__ g2,
    const float* __restrict__ gamma, const float* __restrict__ beta,
    float* __restrict__ y) {
    __shared__ float red[128];
    int r = blockIdx.x;
    int d = threadIdx.x;
    float xv = xp[(size_t)r * DIN + d];
    float gv = g2[(size_t)r * DIN + d];
    if (gv != gv) gv = xv;                 // NaN -> xp
    float o = xv + gv;
    red[d] = o;
    __syncthreads();
    for (int off = 64; off; off >>= 1) {
        if (d < off) red[d] += red[d + off];
        __syncthreads();
    }
    float mu = red[0] * (1.0f / DIN);
    __syncthreads();
    float dv = o - mu;
    red[d] = dv * dv;
    __syncthreads();
    for (int off = 64; off; off >>= 1) {
        if (d < off) red[d] += red[d + off];
        __syncthreads();
    }
    float var = red[0] * (1.0f / DIN);
    y[(size_t)r * DIN + d] = dv * rsqrtf(var + LN_EPS) * gamma[d] + beta[d];
}

extern "C" void kernel_launch(void* const* d_in, const int* in_sizes, int n_in,
                              void* d_out, int out_size, void* d_ws, size_t ws_size,
                              hipStream_t stream) {
    const float* x     = (const float*)d_in[0];
    const int*   src   = (const int*)d_in[1];
    const int*   dst   = (const int*)d_in[2];
    const float* W1s   = (const float*)d_in[3];
    const float* b1s   = (const float*)d_in[4];
    const float* W1d   = (const float*)d_in[5];
    const float* b1d   = (const float*)d_in[6];
    const float* attn1 = (const float*)d_in[7];
    const float* W2s   = (const float*)d_in[8];
    const float* b2s   = (const float*)d_in[9];
    const float* W2d   = (const float*)d_in[10];
    const float* b2d   = (const float*)d_in[11];
    const float* attn2 = (const float*)d_in[12];
    const float* gamma = (const float*)d_in[13];
    const float* beta  = (const float*)d_in[14];
    const float* Wfc   = (const float*)d_in[15];
    const float* bfc   = (const float*)d_in[16];
    float* out = (float*)d_out;

    char* ws = (char*)d_ws;
    const size_t MB = 1024 * 1024;
    float* xp   = (float*)(ws + 0);             // 16 MB, live to the end
    float* fs1  = (float*)(ws + 16 * MB);       // 64 MB; later fs2/fd2/g2
    float* fd1  = (float*)(ws + 80 * MB);       // 64 MB; later g1, later y_ln
    float* ex1  = (float*)(ws + 144 * MB);      // 4 MB
    float* den1 = (float*)(ws + 148 * MB);      // 512 KB
    float* ex2  = (float*)(ws + 149 * MB);      // 1 MB
    float* den2 = (float*)(ws + 150 * MB);      // 128 KB
    int* rowptr = (int*)(ws + 151 * MB);        // (N+1) ints
    int* eidx   = (int*)(ws + 151 * MB + 4096); // E ints
    float* g1  = fd1;                           // reuse fd1 after logits1
    float* fs2 = fs1;                           // reuse fs1 after agg1
    float* fd2 = fs1 + 4 * 1024 * 1024;         // +16 MB
    float* g2  = fs1 + 8 * 1024 * 1024;         // +32 MB
    float* yln = fd1;                           // reuse g1 after fd2 GEMM

    const int M = N_NODES * SEQ;                // 32768 rows, row = n*SEQ + s

    k_preprocess<<<(M * DIN) / 256, 256, 0, stream>>>(x, xp, M * DIN);
    k_build_csr<<<1, N_NODES, 0, stream>>>(dst, rowptr, eidx);

    // layer 1 projections: [32768,128] @ [128,512]  (MT=4 -> 32 KB LDS tile)
    k_gemm_wmma<4><<<dim3(M / 64, 4), 256, 64 * DIN * 4, stream>>>(
        xp, W1s, b1s, fs1, M, DIN, HEADS * DIN);
    k_gemm_wmma<4><<<dim3(M / 64, 4), 256, 64 * DIN * 4, stream>>>(
        xp, W1d, b1d, fd1, M, DIN, HEADS * DIN);

    k_logits1<<<(NEDGE * SEQ) / 8, 256, 0, stream>>>(fs1, fd1, src, dst, attn1, ex1);
    k_den1<<<(N_NODES * SEQ * HEADS) / 256, 256, 0, stream>>>(ex1, rowptr, eidx, den1);
    k_agg1<<<M, 256, 0, stream>>>(fs1, ex1, den1, src, rowptr, eidx, g1);

    // layer 2 projections: [32768,512] @ [512,128]  (MT=2 -> 64 KB LDS tile)
    k_gemm_wmma<2><<<dim3(M / 32, 1), 256, 32 * HEADS * DIN * 4, stream>>>(
        g1, W2s, b2s, fs2, M, HEADS * DIN, DIN);
    k_gemm_wmma<2><<<dim3(M / 32, 1), 256, 32 * HEADS * DIN * 4, stream>>>(
        g1, W2d, b2d, fd2, M, HEADS * DIN, DIN);

    k_logits2<<<(NEDGE * SEQ) / 8, 256, 0, stream>>>(fs2, fd2, src, dst, attn2, ex2);
    k_den2<<<(N_NODES * SEQ) / 256, 256, 0, stream>>>(ex2, rowptr, eidx, den2);
    k_agg2<<<M, DIN, 0, stream>>>(fs2, ex2, den2, src, rowptr, eidx, g2);

    k_res_ln<<<M, DIN, 0, stream>>>(xp, g2, gamma, beta, yln);

    // final fc: [32768,128] @ [128,128] -> d_out  (MT=4)
    k_gemm_wmma<4><<<dim3(M / 64, 1), 256, 64 * DIN * 4, stream>>>(
        yln, Wfc, bfc, out, M, DIN, DOUT);
}